// RegionLoss_3882650436557
// MI455X (gfx1250) — compile-verified
//
#include <hip/hip_runtime.h>
#include <hip/hip_bf16.h>
#include <math.h>

// ---------------- problem constants (from reference) ----------------
#define NB 64
#define NA 5
#define NC 8
#define NH 64
#define NW 64
#define NT 50
#define CH 15          // 7 + NUM_CLASSES channels per anchor
#define ENT 16         // floats per precomputed target record
#define IGNORE_THRES 0.6f
#define PLANE (NH*NW)  // 4096

typedef __attribute__((ext_vector_type(2))) float v2f;
typedef __attribute__((ext_vector_type(8))) float v8f;

// ---------------------------------------------------------------
// Kernel 1: per-(batch,target) precompute. One thread per (b,t).
// Record layout (16 floats):
//  [0]=gi [1]=gj [2..6]=iou[0..4] [7]=best_n [8]=tx [9]=ty [10]=tw [11]=th
//  [12]=label [13]=valid [14,15]=pad
// ---------------------------------------------------------------
__global__ void targets_kernel(const float* __restrict__ tgt,
                               float* __restrict__ table) {
  int tid = blockIdx.x * blockDim.x + threadIdx.x;
  if (tid >= NB * NT) return;
  const float aw[NA] = {1.08f, 3.42f, 6.63f, 9.42f, 16.62f};
  const float ah[NA] = {1.19f, 4.41f, 11.38f, 5.11f, 10.52f};

  const float* p = tgt + (size_t)tid * 5;
  float cls = p[0], c1 = p[1], c2 = p[2], c3 = p[3], c4 = p[4];
  float ssum = cls + c1 + c2 + c3 + c4;
  float valid = (ssum != 0.0f) ? 1.0f : 0.0f;

  float gx = c1 * (float)NW, gy = c2 * (float)NH;
  float gw = c3 * (float)NW, gl = c4 * (float)NH;
  int gi = (int)gx; gi = gi < 0 ? 0 : (gi > NW - 1 ? NW - 1 : gi);
  int gj = (int)gy; gj = gj < 0 ? 0 : (gj > NH - 1 ? NH - 1 : gj);

  float a1 = (gw + 1.0f) * (gl + 1.0f);
  float iou[NA];
  int bn = 0;
  float best = -1e30f;
#pragma unroll
  for (int k = 0; k < NA; ++k) {
    float iw = fmaxf(fminf(gw, aw[k]) + 1.0f, 0.0f);
    float ih = fmaxf(fminf(gl, ah[k]) + 1.0f, 0.0f);
    float inter = iw * ih;
    float a2 = (aw[k] + 1.0f) * (ah[k] + 1.0f);
    iou[k] = inter / (a1 + a2 - inter + 1e-16f);
    if (iou[k] > best) { best = iou[k]; bn = k; }  // first-max like argmax
  }

  float txv = gx - (float)gi;
  float tyv = gy - (float)gj;
  float twv = logf(gw / aw[bn] + 1e-16f);
  float thv = logf(gl / ah[bn] + 1e-16f);
  int lab = (int)cls; lab = lab < 0 ? 0 : (lab > NC - 1 ? NC - 1 : lab);

  float* e = table + (size_t)tid * ENT;
  e[0] = (float)gi; e[1] = (float)gj;
#pragma unroll
  for (int k = 0; k < NA; ++k) e[2 + k] = iou[k];
  e[7] = (float)bn;
  e[8] = txv; e[9] = tyv; e[10] = twv; e[11] = thv;
  e[12] = (float)lab; e[13] = valid; e[14] = 0.0f; e[15] = 0.0f;
}

__device__ __forceinline__ float sigmoidf_(float z) {
  return 1.0f / (1.0f + __expf(-z));
}

// ---------------------------------------------------------------
// Kernel 2: one block per (b,a) plane (320 blocks x 256 threads).
// Replays the sequential scan per cell (last-writer-wins), reads the
// conf plane densely and the other 12 channels only where mask==1.
// 9 deterministic block partials -> ws.
// Accumulator columns: 0=sum(m) 1=lx 2=ly 3=lw 4=lh 5=sum(cmf)
//                      6=sum(cmf*bce) 7=sum(m*bce) 8=sum(m*(lse-picked))
// ---------------------------------------------------------------
__global__ __launch_bounds__(256) void cell_kernel(const float* __restrict__ x,
                                                   const float* __restrict__ table,
                                                   float* __restrict__ partials) {
  __shared__ float tab[NT * ENT];      // 3.2 KB
  __shared__ float red[256][10];       // 10 KB (padded row to dodge conflicts)

  int tid = threadIdx.x;
  int b = blockIdx.x / NA;
  int a = blockIdx.x % NA;

  for (int i0 = tid; i0 < NT * ENT; i0 += 256)
    tab[i0] = table[(size_t)b * NT * ENT + i0];
  __syncthreads();

  const float* xb = x + ((size_t)b * NA * CH + (size_t)a * CH) * PLANE;

  float acc[9];
#pragma unroll
  for (int k = 0; k < 9; ++k) acc[k] = 0.0f;

  for (int c16 = 0; c16 < 16; ++c16) {
    int cell = tid + 256 * c16;            // 0..4095
    int j = cell >> 6, i = cell & 63;
    float fi = (float)i, fj = (float)j;

    // replay the t=0..49 sequential scan for this cell
    float m = 0.0f, cm = 1.0f;
    float tx = 0.0f, ty = 0.0f, tw = 0.0f, th = 0.0f;
    int tlab = 0;
    for (int t = 0; t < NT; ++t) {
      const float* e = &tab[t * ENT];
      if (e[13] == 0.0f) continue;         // uniform per block (b fixed)
      if (e[0] == fi && e[1] == fj) {
        if (e[2 + a] > IGNORE_THRES) cm = 0.0f;   // ignore pass first
        if ((int)e[7] == a) {                      // then assignment pass
          m = 1.0f; cm = 1.0f;
          tx = e[8]; ty = e[9]; tw = e[10]; th = e[11];
          tlab = (int)e[12];
        }
      }
    }

    int off = j * NW + i;
    // dense: conf channel only
    float pconf = sigmoidf_(xb[6 * PLANE + off]);
    float bce = (m != 0.0f) ? -__logf(fmaxf(pconf, 1e-12f))
                            : -__logf(fmaxf(1.0f - pconf, 1e-12f));
    float cmf = cm * (1.0f - m);
    acc[5] += cmf;
    acc[6] += cmf * bce;
    acc[7] += m * bce;

    if (m != 0.0f) {   // sparse: <=50 cells per batch ever take this path
      acc[0] += 1.0f;
      float px = sigmoidf_(xb[0 * PLANE + off]);
      float py = sigmoidf_(xb[1 * PLANE + off]);
      float pw = xb[2 * PLANE + off];
      float ph = xb[3 * PLANE + off];
      acc[1] += (px - tx) * (px - tx);
      acc[2] += (py - ty) * (py - ty);
      acc[3] += (pw - tw) * (pw - tw);
      acc[4] += (ph - th) * (ph - th);
      float sc[NC]; float mx = -1e30f;
#pragma unroll
      for (int c = 0; c < NC; ++c) {
        sc[c] = sigmoidf_(xb[(7 + c) * PLANE + off]);
        mx = fmaxf(mx, sc[c]);
      }
      float se = 0.0f;
#pragma unroll
      for (int c = 0; c < NC; ++c) se += __expf(sc[c] - mx);
      float lse = __logf(se) + mx;
      acc[8] += lse - sc[tlab];
    }
  }

#pragma unroll
  for (int k = 0; k < 9; ++k) red[tid][k] = acc[k];
  __syncthreads();
  for (int s = 128; s > 0; s >>= 1) {
    if (tid < s) {
#pragma unroll
      for (int k = 0; k < 9; ++k) red[tid][k] += red[tid + s][k];
    }
    __syncthreads();
  }
  if (tid < 9) partials[(size_t)blockIdx.x * 9 + tid] = red[0][tid];
}

// ---------------------------------------------------------------
// Kernel 3: single-wave WMMA reduction of R x 9 partials.
// D = ones(16x4) x B(4x16) + C : each v_wmma_f32_16x16x4_f32 folds
// 4 partial records into the accumulator columns; column j of D row 0
// ends up holding total[j]. Two C accumulators break the WMMA chain.
// EXEC is all ones (32 threads, no divergence) per ISA requirement.
// ---------------------------------------------------------------
__global__ __launch_bounds__(32) void finalize_kernel(const float* __restrict__ partials,
                                                      float* __restrict__ out,
                                                      int R) {
  int lane = threadIdx.x;        // 0..31
  int n = lane & 15;             // B column
  int kh = lane >> 4;            // which K-half this lane supplies
  int cn = (n < 9) ? n : 8;      // clamp address; value zeroed below

  v2f a; a.x = 1.0f; a.y = 1.0f; // A = ones(16x4), layout-invariant
  v8f c0 = {}; v8f c1 = {};

  for (int r0 = 0; r0 + 8 <= R; r0 += 8) {
    int ra = r0 + 2 * kh;                // rows ra, ra+1 -> K {2kh, 2kh+1}
    float v0 = partials[ra * 9 + cn];
    float v1 = partials[(ra + 1) * 9 + cn];
    v2f bA; bA.x = (n < 9) ? v0 : 0.0f; bA.y = (n < 9) ? v1 : 0.0f;
    c0 = __builtin_amdgcn_wmma_f32_16x16x4_f32(
        false, a, false, bA, (short)0, c0, false, false);

    int rb = r0 + 4 + 2 * kh;
    float w0 = partials[rb * 9 + cn];
    float w1 = partials[(rb + 1) * 9 + cn];
    v2f bB; bB.x = (n < 9) ? w0 : 0.0f; bB.y = (n < 9) ? w1 : 0.0f;
    c1 = __builtin_amdgcn_wmma_f32_16x16x4_f32(
        false, a, false, bB, (short)0, c1, false, false);
  }

  v8f c = c0 + c1;
  float tot = c[0];              // lane L holds total[L & 15] (all D rows equal)

  float S[9];
#pragma unroll
  for (int jj = 0; jj < 9; ++jj) S[jj] = __shfl(tot, jj, 32);

  if (lane == 0) {
    float cnt  = fmaxf(S[0], 1.0f);
    float cntf = fmaxf(S[5], 1.0f);
    float loss = (S[1] + S[2] + S[3] + S[4]) / cnt   // x,y,w,h
               + S[6] / cntf + S[7] / cnt            // conf
               + (1.0f / (float)NB) * (S[8] / cnt);  // cls
    out[0] = loss;
  }
}

// ---------------------------------------------------------------
extern "C" void kernel_launch(void* const* d_in, const int* in_sizes, int n_in,
                              void* d_out, int out_size, void* d_ws, size_t ws_size,
                              hipStream_t stream) {
  const float* x   = (const float*)d_in[0];   // (64, 75, 64, 64) fp32
  const float* tgt = (const float*)d_in[1];   // (64, 50, 5) fp32
  float* out = (float*)d_out;

  float* table    = (float*)d_ws;                       // 3200*16 floats
  float* partials = table + (size_t)NB * NT * ENT;      // 320*9 floats

  targets_kernel<<<(NB * NT + 255) / 256, 256, 0, stream>>>(tgt, table);
  cell_kernel<<<NB * NA, 256, 0, stream>>>(x, table, partials);
  finalize_kernel<<<1, 32, 0, stream>>>(partials, out, NB * NA);
}